// OnlineTripletLoss_65558380806266
// MI455X (gfx1250) — compile-verified
//
#include <hip/hip_runtime.h>
#include <hip/hip_bf16.h>
#include <math.h>

typedef __attribute__((ext_vector_type(2))) float v2f;
typedef __attribute__((ext_vector_type(8))) float v8f;
typedef unsigned long long u64;

#define TL_EPS    1e-6f
#define TL_MARGIN 1.0f
#define TL_D      64
#define MINE_WAVES 4
#define COL_SPLIT  4

// ---------------------------------------------------------------------------
// Kernel 1: per-row stats  sq[i] = ||E[i]||^2,  s[i] = sum(E[i])
// ---------------------------------------------------------------------------
__global__ __launch_bounds__(256) void tl_rowstats(const float* __restrict__ E,
                                                   float* __restrict__ sq,
                                                   float* __restrict__ s, int Bn) {
  int i = blockIdx.x * blockDim.x + threadIdx.x;
  if (i >= Bn) return;
  const float* e = E + (size_t)i * TL_D;
  float a = 0.f, b = 0.f;
#pragma unroll
  for (int k = 0; k < TL_D; ++k) { float x = e[k]; a = fmaf(x, x, a); b += x; }
  sq[i] = a;
  s[i] = b;
}

// Init: per-row merge keys + global accumulators.
__global__ __launch_bounds__(256) void tl_init(u64* __restrict__ maxKey,
                                               u64* __restrict__ minKey,
                                               float* __restrict__ acc, int Bn) {
  int i = blockIdx.x * blockDim.x + threadIdx.x;
  if (i < Bn) {
    maxKey[i] = 0ull;
    minKey[i] = ~0ull;
  }
  if (i == 0) { acc[0] = 0.f; acc[1] = 0.f; }
}

// Monotone (d2, idx) packing: d2 >= 0 so its f32 bit pattern orders correctly
// as unsigned. Max side complements idx so ties pick the smallest index
// (matching argmax-first semantics); min side keeps idx directly.
__device__ __forceinline__ u64 tl_pack_max(float d2, int idx) {
  return ((u64)__float_as_uint(d2) << 32) | (unsigned int)(~idx);
}
__device__ __forceinline__ u64 tl_pack_min(float d2, int idx) {
  return ((u64)__float_as_uint(d2) << 32) | (unsigned int)idx;
}

// ---------------------------------------------------------------------------
// Mining helpers
// ---------------------------------------------------------------------------
__device__ __forceinline__ void tl_load_btile(
    const float* __restrict__ E, const float* __restrict__ sq,
    const float* __restrict__ s, const int* __restrict__ lab,
    int colbase, int l16, int k0,
    v2f (&frag)[16], float& colSq, float& colS, int& colLab, int& cbOut) {
  const int cb = colbase + l16;
  const float* Eb = E + (size_t)cb * TL_D;
  __builtin_prefetch(Eb + 64 * TL_D, 0, 3);   // ~4 tiles ahead
#pragma unroll
  for (int t = 0; t < 16; ++t) {
    frag[t].x = Eb[4 * t + k0];
    frag[t].y = Eb[4 * t + k0 + 1];
  }
  colSq  = sq[cb];
  colS   = s[cb];
  colLab = lab[cb];
  cbOut  = cb;
}

__device__ __forceinline__ void tl_compute_tile(
    const v2f (&afrag)[16], const v2f (&bfrag)[16],
    float colSq, float colS, int colLab, int cb,
    const float (&rowSq)[8], const float (&rowS)[8], const int (&rowLab)[8],
    float (&maxV)[8], int (&maxI)[8], float (&minV)[8], int (&minI)[8]) {
  // Two independent accumulator chains (even/odd K-steps) for WMMA-pipe ILP.
  v8f acc0 = {}, acc1 = {};
#pragma unroll
  for (int t = 0; t < 8; ++t) {
    acc0 = __builtin_amdgcn_wmma_f32_16x16x4_f32(
        false, afrag[2 * t], false, bfrag[2 * t], (short)0, acc0, false, false);
    acc1 = __builtin_amdgcn_wmma_f32_16x16x4_f32(
        false, afrag[2 * t + 1], false, bfrag[2 * t + 1], (short)0, acc1, false, false);
  }
  const float epsConst = (float)TL_D * TL_EPS * TL_EPS;
  // Branchless running argmax (positives) / argmin (negatives): pure cndmask.
#pragma unroll
  for (int v = 0; v < 8; ++v) {
    float g = acc0[v] + acc1[v];
    float d2 = rowSq[v] + colSq - 2.0f * g
             + 2.0f * TL_EPS * (rowS[v] - colS) + epsConst;
    bool pos = (colLab == rowLab[v]);
    bool uM = pos && (d2 > maxV[v]);
    maxV[v] = uM ? d2 : maxV[v];
    maxI[v] = uM ? cb : maxI[v];
    bool uN = (!pos) && (d2 < minV[v]);
    minV[v] = uN ? d2 : minV[v];
    minI[v] = uN ? cb : minI[v];
  }
}

// ---------------------------------------------------------------------------
// Kernel 2: fused gram-GEMM + batch-hard mining.
// grid.x picks a 16-row anchor tile group (MINE_WAVES tiles per block),
// grid.y splits the column range COL_SPLIT ways for machine fill.
// Per-wave partial argmax/argmin merged across column chunks with u64 atomics.
// ---------------------------------------------------------------------------
__global__ __launch_bounds__(32 * MINE_WAVES) void tl_mine(
    const float* __restrict__ E, const int* __restrict__ lab,
    const float* __restrict__ sq, const float* __restrict__ s,
    u64* __restrict__ maxKey, u64* __restrict__ minKey, int Bn) {
  const int wave = threadIdx.x >> 5;
  const int lane = threadIdx.x & 31;
  const int half = lane >> 4;   // 0: lanes 0-15, 1: lanes 16-31
  const int l16  = lane & 15;
  const int k0   = half << 1;   // K base for A/B fragments

  const int rowbase = (blockIdx.x * MINE_WAVES + wave) << 4;
  if (rowbase >= Bn) return;    // wave-uniform; EXEC stays all-ones for WMMA

  // This wave's column-tile range.
  const int nTiles = Bn >> 4;
  const int chunk  = (nTiles + gridDim.y - 1) / gridDim.y;
  const int tile0  = blockIdx.y * chunk;
  const int tile1  = min(tile0 + chunk, nTiles);
  if (tile0 >= nTiles) return;  // uniform

  // Preload all 16 A K-step fragments for this wave's 16 rows.
  const int ra = rowbase + l16;
  const float* Ea = E + (size_t)ra * TL_D;
  v2f afrag[16];
#pragma unroll
  for (int t = 0; t < 16; ++t) {
    afrag[t].x = Ea[4 * t + k0];
    afrag[t].y = Ea[4 * t + k0 + 1];
  }

  // Per-slot (row) constants: slot v maps to row rowbase + half*8 + v.
  float rowSq[8], rowS[8];
  int rowLab[8];
#pragma unroll
  for (int v = 0; v < 8; ++v) {
    int r = rowbase + (half << 3) + v;
    rowSq[v]  = sq[r];
    rowS[v]   = s[r];
    rowLab[v] = lab[r];
  }

  float maxV[8], minV[8];
  int   maxI[8], minI[8];
#pragma unroll
  for (int v = 0; v < 8; ++v) {
    maxV[v] = -3.0e38f; minV[v] = 3.0e38f; maxI[v] = 0; minI[v] = 0;
  }

  v2f bufA[16], bufB[16];
  float sqA, sA, sqB, sB;
  int labA, labB, cbA, cbB;

  tl_load_btile(E, sq, s, lab, tile0 << 4, l16, k0, bufA, sqA, sA, labA, cbA);

  // Unrolled-by-2 register ping-pong. Tail clamping re-processes the last
  // tile, which is idempotent for running max/min.
  for (int tile = tile0; tile < tile1; tile += 2) {
    int nb1 = ((tile + 1 < tile1) ? tile + 1 : tile1 - 1) << 4;
    tl_load_btile(E, sq, s, lab, nb1, l16, k0, bufB, sqB, sB, labB, cbB);
    tl_compute_tile(afrag, bufA, sqA, sA, labA, cbA,
                    rowSq, rowS, rowLab, maxV, maxI, minV, minI);

    int nb2 = ((tile + 2 < tile1) ? tile + 2 : tile1 - 1) << 4;
    tl_load_btile(E, sq, s, lab, nb2, l16, k0, bufA, sqA, sA, labA, cbA);
    tl_compute_tile(afrag, bufB, sqB, sB, labB, cbB,
                    rowSq, rowS, rowLab, maxV, maxI, minV, minI);
  }

  // Cross-lane argmax/argmin within each 16-lane half (xor masks <= 8 never
  // cross the half boundary on wave32), then cross-chunk merge via u64 atomics.
#pragma unroll
  for (int v = 0; v < 8; ++v) {
    float mv = maxV[v]; int mi = maxI[v];
    float nv = minV[v]; int ni = minI[v];
#pragma unroll
    for (int off = 8; off >= 1; off >>= 1) {
      float omv = __shfl_xor(mv, off);
      int   omi = __shfl_xor(mi, off);
      bool tM = (omv > mv) || (omv == mv && omi < mi);
      mv = tM ? omv : mv;  mi = tM ? omi : mi;
      float onv = __shfl_xor(nv, off);
      int   oni = __shfl_xor(ni, off);
      bool tN = (onv < nv) || (onv == nv && oni < ni);
      nv = tN ? onv : nv;  ni = tN ? oni : ni;
    }
    if (l16 == 0) {
      int r = rowbase + (half << 3) + v;
      if (mv > -2.9e38f) atomicMax(&maxKey[r], tl_pack_max(fmaxf(mv, 0.f), mi));
      if (nv < 2.9e38f)  atomicMin(&minKey[r], tl_pack_min(fmaxf(nv, 0.f), ni));
    }
  }
}

// ---------------------------------------------------------------------------
// Kernel 3: per-row triplet loss + wave-reduced atomic accumulation
// ---------------------------------------------------------------------------
__global__ __launch_bounds__(256) void tl_loss(
    const float* __restrict__ E, const u64* __restrict__ maxKey,
    const u64* __restrict__ minKey, float* __restrict__ acc, int Bn) {
  int i = blockIdx.x * blockDim.x + threadIdx.x;
  float per = 0.f, cnt = 0.f;
  if (i < Bn) {
    u64 mk = maxKey[i];
    u64 nk = minKey[i];
    bool vld = (mk != 0ull) && (nk != ~0ull);
    if (vld) {
      int hpI = (int)(~(unsigned int)mk);
      int hnI = (int)((unsigned int)nk);
      const float* a = E + (size_t)i * TL_D;
      const float* p = E + (size_t)hpI * TL_D;
      const float* n = E + (size_t)hnI * TL_D;
      float ap = 0.f, an = 0.f, pn = 0.f;
#pragma unroll
      for (int k = 0; k < TL_D; ++k) {
        float da = a[k] - p[k] + TL_EPS; ap = fmaf(da, da, ap);
        float db = a[k] - n[k] + TL_EPS; an = fmaf(db, db, an);
        float dc = p[k] - n[k] + TL_EPS; pn = fmaf(dc, dc, pn);
      }
      ap = sqrtf(ap); an = sqrtf(an); pn = sqrtf(pn);
      per = fmaxf(ap - fminf(an, pn) + TL_MARGIN, 0.f);
      cnt = 1.f;
    }
  }
#pragma unroll
  for (int off = 16; off >= 1; off >>= 1) {
    per += __shfl_xor(per, off);
    cnt += __shfl_xor(cnt, off);
  }
  if ((threadIdx.x & 31) == 0) {
    atomicAdd(&acc[0], per);
    atomicAdd(&acc[1], cnt);
  }
}

__global__ void tl_final(const float* __restrict__ acc, float* __restrict__ out) {
  if (threadIdx.x == 0) out[0] = acc[0] / fmaxf(acc[1], 1.0f);
}

// ---------------------------------------------------------------------------
// Host launch
// ---------------------------------------------------------------------------
extern "C" void kernel_launch(void* const* d_in, const int* in_sizes, int n_in,
                              void* d_out, int out_size, void* d_ws, size_t ws_size,
                              hipStream_t stream) {
  (void)n_in; (void)out_size; (void)ws_size;
  const float* E   = (const float*)d_in[0];
  const int*   lab = (const int*)d_in[1];
  // d_in[2]/d_in[3] (positive_idxs/negative_idxs) are label-derived; unused.
  const int Bn = in_sizes[1];

  // Workspace layout (u64 arrays first for 8B alignment).
  u64* maxKey = (u64*)d_ws;
  u64* minKey = maxKey + Bn;
  float* f    = (float*)(minKey + Bn);
  float* sq   = f;
  float* s    = f + Bn;
  float* acc  = f + 2 * (size_t)Bn;
  float* out  = (float*)d_out;

  tl_rowstats<<<(Bn + 255) / 256, 256, 0, stream>>>(E, sq, s, Bn);
  tl_init<<<(Bn + 255) / 256, 256, 0, stream>>>(maxKey, minKey, acc, Bn);

  const int rowsPerBlock = 16 * MINE_WAVES;
  dim3 mgrid((Bn + rowsPerBlock - 1) / rowsPerBlock, COL_SPLIT);
  dim3 mblk(32 * MINE_WAVES);
  tl_mine<<<mgrid, mblk, 0, stream>>>(E, lab, sq, s, maxKey, minKey, Bn);

  tl_loss<<<(Bn + 255) / 256, 256, 0, stream>>>(E, maxKey, minKey, acc, Bn);
  tl_final<<<1, 32, 0, stream>>>(acc, out);
}